// KapoorConv_79474074845505
// MI455X (gfx1250) — compile-verified
//
#include <hip/hip_runtime.h>

// KapoorConv (2-layer GCN, D=64) for MI455X / gfx1250, wave32.
// Dense transforms use V_WMMA_F32_16X16X4_F32 (fp32 matrix pipe, no precision
// loss, the GEMM is only ~1.6 GFLOP). The sparse gather/scatter is L2-resident
// (h = 25.6MB << 192MB L2) and uses non-returning global_atomic_add_f32.

typedef __attribute__((ext_vector_type(2))) float v2f;
typedef __attribute__((ext_vector_type(8))) float v8f;

#define DIM 64
#define LDS_STRIDE 66  // even (keeps 8B-aligned ds_load_b64), padded vs 64 for banks

// ---------------- degree / normalization ----------------

__global__ void k_deg_init(float* __restrict__ deg, int n) {
  int i = blockIdx.x * blockDim.x + threadIdx.x;
  if (i < n) deg[i] = 1.0f;  // self-loop
}

__global__ void k_deg_count(const int* __restrict__ dst, float* __restrict__ deg, int E) {
  int e = blockIdx.x * blockDim.x + threadIdx.x;
  if (e < E) atomicAdd(&deg[dst[e]], 1.0f);  // no-return global_atomic_add_f32
}

__global__ void k_deg_rsqrt(float* __restrict__ deg, int n) {
  int i = blockIdx.x * blockDim.x + threadIdx.x;
  if (i < n) deg[i] = rsqrtf(deg[i]);  // deg >= 1
}

// ---------------- O[N x 64] = act(A[N x 64]) @ W[64 x 64] via WMMA ----------------
// One wave per 16-row block; 4 v8f accumulators cover the full 16x64 output.
// nRows must be a multiple of 16 (100000 = 6250*16) so EXEC is all-ones for WMMA.
// RELU is a compile-time template parameter so layer 1 carries zero activation
// instructions and layer 2 gets a bare v_max_num_f32 (no if-conversion selects).

template <bool RELU>
__global__ void __launch_bounds__(128) k_gemm64(const float* __restrict__ A,
                                                const float* __restrict__ W,
                                                float* __restrict__ O,
                                                int nRows) {
  __shared__ float sWT[DIM * LDS_STRIDE];  // W transposed: sWT[col*66 + row]

  // Stage W into LDS transposed (coalesced global read, one-time scatter write).
  for (int t = threadIdx.x; t < DIM * DIM; t += blockDim.x) {
    int r = t >> 6, c = t & 63;
    sWT[c * LDS_STRIDE + r] = W[t];
  }
  __syncthreads();

  const int lane = threadIdx.x & 31;
  const int wave = threadIdx.x >> 5;
  const int rowBlock = blockIdx.x * (blockDim.x >> 5) + wave;
  const int rowBase = rowBlock * 16;
  if (rowBase >= nRows) return;

  const int half = lane >> 4;  // 0: K={0,1}(+4k), 1: K={2,3}(+4k)
  const int l15 = lane & 15;

  // A fragment base: row = rowBase + l15, cols (4k + 2*half, 4k + 2*half + 1)
  const float* aBase = A + (size_t)(rowBase + l15) * DIM + 2 * half;
  // B fragment base (transposed W in LDS): col = 16n + l15, rows (4k+2*half, +1)
  const float* wBase = sWT + l15 * LDS_STRIDE + 2 * half;

  v8f acc0 = {}, acc1 = {}, acc2 = {}, acc3 = {};

#pragma unroll
  for (int k = 0; k < 16; ++k) {
    v2f a = *(const v2f*)(aBase + 4 * k);
    if constexpr (RELU) {
      a.x = fmaxf(a.x, 0.0f);
      a.y = fmaxf(a.y, 0.0f);
    }
    v2f b0 = *(const v2f*)(wBase + 0 * 16 * LDS_STRIDE + 4 * k);
    v2f b1 = *(const v2f*)(wBase + 1 * 16 * LDS_STRIDE + 4 * k);
    v2f b2 = *(const v2f*)(wBase + 2 * 16 * LDS_STRIDE + 4 * k);
    v2f b3 = *(const v2f*)(wBase + 3 * 16 * LDS_STRIDE + 4 * k);
    acc0 = __builtin_amdgcn_wmma_f32_16x16x4_f32(false, a, false, b0, (short)0, acc0, false, false);
    acc1 = __builtin_amdgcn_wmma_f32_16x16x4_f32(false, a, false, b1, (short)0, acc1, false, false);
    acc2 = __builtin_amdgcn_wmma_f32_16x16x4_f32(false, a, false, b2, (short)0, acc2, false, false);
    acc3 = __builtin_amdgcn_wmma_f32_16x16x4_f32(false, a, false, b3, (short)0, acc3, false, false);
  }

  // C/D layout: VGPR v -> row v (lanes 0-15) / row v+8 (lanes 16-31), col = l15.
#pragma unroll
  for (int v = 0; v < 8; ++v) {
    float* o = O + (size_t)(rowBase + v + 8 * half) * DIM + l15;
    o[0]  = acc0[v];
    o[16] = acc1[v];
    o[32] = acc2[v];
    o[48] = acc3[v];
  }
}

// ---------------- self-loop contribution (+ optional bias); initializes out ----------------

__global__ void k_selfloop_init(const float* __restrict__ h, const float* __restrict__ dinv,
                                const float* __restrict__ bias, float* __restrict__ out, int n) {
  int idx = blockIdx.x * blockDim.x + threadIdx.x;  // over n*16 (float4 each)
  if (idx >= n * 16) return;
  int i = idx >> 4;
  int d = (idx & 15) * 4;
  float s = dinv[i];
  s = s * s;  // self-loop norm = dinv[i]^2
  float4 hv = *(const float4*)(h + (size_t)i * DIM + d);
  float4 o;
  o.x = hv.x * s; o.y = hv.y * s; o.z = hv.z * s; o.w = hv.w * s;
  if (bias) {
    float4 bv = *(const float4*)(bias + d);
    o.x += bv.x; o.y += bv.y; o.z += bv.z; o.w += bv.w;
  }
  *(float4*)(out + (size_t)i * DIM + d) = o;
}

// ---------------- edge scatter: out[i,:] += h[j,:] * dinv[i]*dinv[j] ----------------

__global__ void k_agg_edges(const float* __restrict__ h, const float* __restrict__ dinv,
                            const int* __restrict__ src, const int* __restrict__ dst,
                            float* __restrict__ out, int E) {
  long idx = (long)blockIdx.x * blockDim.x + threadIdx.x;  // over E*16 (float4 each)
  if (idx >= (long)E * 16) return;
  int e = (int)(idx >> 4);
  int d = ((int)idx & 15) * 4;
  int j = src[e];
  int i = dst[e];
  float nrm = dinv[i] * dinv[j];
  float4 hv = *(const float4*)(h + (size_t)j * DIM + d);
  float* o = out + (size_t)i * DIM + d;
  atomicAdd(o + 0, hv.x * nrm);
  atomicAdd(o + 1, hv.y * nrm);
  atomicAdd(o + 2, hv.z * nrm);
  atomicAdd(o + 3, hv.w * nrm);
}

// ---------------- launcher ----------------

extern "C" void kernel_launch(void* const* d_in, const int* in_sizes, int n_in,
                              void* d_out, int out_size, void* d_ws, size_t ws_size,
                              hipStream_t stream) {
  const float* x  = (const float*)d_in[0];
  const float* W1 = (const float*)d_in[1];
  const float* W2 = (const float*)d_in[2];
  const float* b2 = (const float*)d_in[3];
  const int*   ei = (const int*)d_in[4];

  const int N = in_sizes[0] / DIM;     // 100000
  const int E = in_sizes[4] / 2;       // 1600000
  const int* src = ei;                 // edge_index[0]
  const int* dst = ei + E;             // edge_index[1]
  float* out = (float*)d_out;

  // Workspace layout: dinv[N] | bufA[N*64] | bufB[N*64]  (~51.6 MB)
  char* ws = (char*)d_ws;
  size_t off = 0;
  float* dinv = (float*)(ws + off); off += (((size_t)N * 4) + 255) & ~(size_t)255;
  float* bufA = (float*)(ws + off); off += (((size_t)N * DIM * 4) + 255) & ~(size_t)255;
  float* bufB = (float*)(ws + off);

  const int T = 256;
  // Normalization: deg -> dinv
  k_deg_init <<<(N + T - 1) / T, T, 0, stream>>>(dinv, N);
  k_deg_count<<<(E + T - 1) / T, T, 0, stream>>>(dst, dinv, E);
  k_deg_rsqrt<<<(N + T - 1) / T, T, 0, stream>>>(dinv, N);

  const int rowBlocks = (N + 15) / 16;         // 6250 (exact)
  const int wavesPerBlock = 4;                 // blockDim = 128
  const int gemmGrid = (rowBlocks + wavesPerBlock - 1) / wavesPerBlock;
  const int nodeVecGrid = (N * 16 + T - 1) / T;
  const unsigned edgeVecGrid = (unsigned)(((long)E * 16 + T - 1) / T);

  // Layer 1: bufA = x @ W1 ; bufB = selfloop + scatter  (ReLU deferred to layer-2 gemm)
  k_gemm64<false><<<gemmGrid, 32 * wavesPerBlock, 0, stream>>>(x, W1, bufA, N);
  k_selfloop_init<<<nodeVecGrid, T, 0, stream>>>(bufA, dinv, nullptr, bufB, N);
  k_agg_edges<<<edgeVecGrid, T, 0, stream>>>(bufA, dinv, src, dst, bufB, E);

  // Layer 2: bufA = relu(bufB) @ W2 ; out = selfloop + bias + scatter
  k_gemm64<true><<<gemmGrid, 32 * wavesPerBlock, 0, stream>>>(bufB, W2, bufA, N);
  k_selfloop_init<<<nodeVecGrid, T, 0, stream>>>(bufA, dinv, b2, out, N);
  k_agg_edges<<<edgeVecGrid, T, 0, stream>>>(bufA, dinv, src, dst, out, E);
}